// PhyloDisentangler_13726715478033
// MI455X (gfx1250) — compile-verified
//
#include <hip/hip_runtime.h>
#include <hip/hip_bf16.h>
#include <math.h>

typedef __attribute__((ext_vector_type(2))) float v2f;
typedef __attribute__((ext_vector_type(8))) float v8f;

#define HW_     1024
#define CH_     256
#define FEATIN_ 131072
#define FEATZ_  512

#define WMMA_F32(A, Bf, C) \
  __builtin_amdgcn_wmma_f32_16x16x4_f32(false, (A), false, (Bf), (short)0, (C), false, false)

__device__ __forceinline__ float silu_f(float x) {
  // x * sigmoid(x); v_exp_f32 + v_rcp_f32, no IEEE-div sequence
  return x * __builtin_amdgcn_rcpf(1.0f + __expf(-x));
}

// out[b][o][p] = sum_c W[o][c] * silu(X[b][c][p]) + bias[o]
// grid (16 n-tiles of 64, 32 batches), block 256 (8 waves).
// Block stages silu(X) panel in LDS with K-pair interleave:
//   lb[(k>>1)*128 + 2n + (k&1)]  ->  each WMMA B-frag is one ds_load_b64.
// Each wave owns two 16-row M tiles (m0, m0+128) x four N tiles -> 8 WMMA/k-step.
__global__ __launch_bounds__(256) void k_conv1x1_silu(
    const float* __restrict__ X, const float* __restrict__ W,
    const float* __restrict__ bias, float* __restrict__ out)
{
  __shared__ float lb[CH_ * 64];              // 64 KB
  const int t  = threadIdx.x;
  const int bz = blockIdx.y;
  const int n0 = blockIdx.x * 64;
  const float* Xb = X + (size_t)bz * (CH_ * HW_);
  for (int i = t; i < CH_ * 64; i += 256) {
    const int k = i >> 6, n = i & 63;
    lb[((k >> 1) << 7) + (n << 1) + (k & 1)] = silu_f(Xb[(size_t)k * HW_ + n0 + n]);
  }
  __syncthreads();

  const int lane = t & 31;
  const int wave = t >> 5;
  const int ln   = lane & 15;
  const int kk   = (lane >> 4) << 1;          // lanes 0-15: K 0,1 ; 16-31: K 2,3
  const int m0   = wave * 16;

  v8f a0c0 = {}, a0c1 = {}, a0c2 = {}, a0c3 = {};
  v8f a1c0 = {}, a1c1 = {}, a1c2 = {}, a1c3 = {};
  const float* w0 = W + (size_t)(m0 + ln) * CH_ + kk;
  const float* w1 = w0 + (size_t)128 * CH_;
  for (int k = 0; k < CH_; k += 4) {
    v2f a0, a1;
    a0.x = w0[k]; a0.y = w0[k + 1];
    a1.x = w1[k]; a1.y = w1[k + 1];
    // (k+kk) even: element offset (k+kk)*64 + 2n ; pair (k,k+1) contiguous
    const float* l0 = lb + (k + kk) * 64 + (ln << 1);
    v2f b0, b1, b2, b3;
    b0.x = l0[0];  b0.y = l0[1];
    b1.x = l0[32]; b1.y = l0[33];
    b2.x = l0[64]; b2.y = l0[65];
    b3.x = l0[96]; b3.y = l0[97];
    a0c0 = WMMA_F32(a0, b0, a0c0);
    a0c1 = WMMA_F32(a0, b1, a0c1);
    a0c2 = WMMA_F32(a0, b2, a0c2);
    a0c3 = WMMA_F32(a0, b3, a0c3);
    a1c0 = WMMA_F32(a1, b0, a1c0);
    a1c1 = WMMA_F32(a1, b1, a1c1);
    a1c2 = WMMA_F32(a1, b2, a1c2);
    a1c3 = WMMA_F32(a1, b3, a1c3);
  }
  const int hi = (lane >> 4) << 3;            // lanes 16-31 hold M = r+8
  float* ob = out + (size_t)bz * (CH_ * HW_);
#pragma unroll
  for (int r = 0; r < 8; ++r) {
    {
      const int m = m0 + hi + r;
      const float bi = bias[m];
      float* o = ob + (size_t)m * HW_ + n0 + ln;
      o[0]  = a0c0[r] + bi;
      o[16] = a0c1[r] + bi;
      o[32] = a0c2[r] + bi;
      o[48] = a0c3[r] + bi;
    }
    {
      const int m = m0 + 128 + hi + r;
      const float bi = bias[m];
      float* o = ob + (size_t)m * HW_ + n0 + ln;
      o[0]  = a1c0[r] + bi;
      o[16] = a1c1[r] + bi;
      o[32] = a1c2[r] + bi;
      o[48] = a1c3[r] + bi;
    }
  }
}

// per-batch layernorm stats over first 128 channels (131072 elems)
__global__ __launch_bounds__(256) void k_ln_stats(const float* __restrict__ h,
                                                  float* __restrict__ mu, float* __restrict__ rs)
{
  const int b = blockIdx.x, t = threadIdx.x;
  const float* p = h + (size_t)b * 262144;
  float s = 0.f, s2 = 0.f;
  for (int i = t; i < FEATIN_; i += 256) { const float v = p[i]; s += v; s2 = fmaf(v, v, s2); }
  __shared__ float as[256], aq[256];
  as[t] = s; aq[t] = s2;
  __syncthreads();
  for (int o = 128; o > 0; o >>= 1) {
    if (t < o) { as[t] += as[t + o]; aq[t] += aq[t + o]; }
    __syncthreads();
  }
  if (t == 0) {
    const float m = as[0] * (1.f / (float)FEATIN_);
    const float var = aq[0] * (1.f / (float)FEATIN_) - m * m;
    mu[b] = m;
    rs[b] = rsqrtf(var + 1e-5f);
  }
}

// hnT[(k>>1)*64 + 2b + (k&1)] = (h[b][k]-mu[b])*rs[b]*ln_w[k] + ln_b[k]
// (transposed + K-pair interleaved for single-b64 WMMA B-frag loads)
__global__ __launch_bounds__(256) void k_ln_apply(
    const float* __restrict__ h, const float* __restrict__ lnw,
    const float* __restrict__ lnb, const float* __restrict__ mu,
    const float* __restrict__ rs, float* __restrict__ hnT)
{
  const int i = blockIdx.x * 256 + threadIdx.x;   // i = k*32 + b
  const int b = i & 31, k = i >> 5;
  const float v = (h[(size_t)b * 262144 + k] - mu[b]) * rs[b] * lnw[k] + lnb[k];
  hnT[(size_t)((k >> 1) << 6) + (b << 1) + (k & 1)] = v;
}

// z.T = W(512x131072) @ hnT(131072x32) via WMMA, split-K.
// grid (32 m-tiles of 16, 64 K-chunks of 2048), block = 1 wave.
// Each weight cacheline is owned by exactly one wave -> 268 MB streamed once.
__global__ __launch_bounds__(32) void k_mlp_in(
    const float* __restrict__ w, const float* __restrict__ hnT,
    float* __restrict__ z_part)
{
  const int lane = threadIdx.x;
  const int ln = lane & 15;
  const int kk = (lane >> 4) << 1;
  const int m0 = blockIdx.x * 16;
  const int k0 = blockIdx.y * 2048;
  const float* wr = w + (size_t)(m0 + ln) * FEATIN_ + kk;
  v8f acc0 = {}, acc1 = {};
  for (int k = k0; k < k0 + 2048; k += 4) {
    v2f a;
    a.x = wr[k]; a.y = wr[k + 1];
    const float* hp = hnT + (size_t)(k + kk) * 32 + (ln << 1);  // (k+kk) even
    v2f b0, b1;
    b0.x = hp[0];  b0.y = hp[1];
    b1.x = hp[32]; b1.y = hp[33];
    acc0 = WMMA_F32(a, b0, acc0);
    acc1 = WMMA_F32(a, b1, acc1);
  }
  const int hi = (lane >> 4) << 3;
  float* zp = z_part + (size_t)blockIdx.y * 16384;   // [n*32 + b]
#pragma unroll
  for (int r = 0; r < 8; ++r) {
    const int m = m0 + hi + r;
    zp[m * 32 + ln]      = acc0[r];
    zp[m * 32 + 16 + ln] = acc1[r];
  }
}

// deterministic fixed-order reduction of the 64 K-chunks + bias -> z_pre[b*512+n]
__global__ __launch_bounds__(256) void k_reduce_z(
    const float* __restrict__ z_part, const float* __restrict__ bias,
    float* __restrict__ z_pre)
{
  const int i = blockIdx.x * 256 + threadIdx.x;   // b*512 + n
  const int b = i >> 9, n = i & 511;
  float s = bias[n];
  for (int c = 0; c < 64; ++c) s += z_part[(size_t)c * 16384 + n * 32 + b];
  z_pre[i] = s;
}

// one block per zf row r = b*32 + cb*4 + l; argmin over 1024 codes (first-min wins).
// Emits zqT in the K-pair-interleaved transposed layout + per-row loss.
__global__ __launch_bounds__(128) void k_vq(const float* __restrict__ z_pre,
                                            const float* __restrict__ cbk,
                                            float* __restrict__ zqT, float* __restrict__ qrow)
{
  const int r = blockIdx.x, t = threadIdx.x;
  const int b = r >> 5, rem = r & 31;
  float zf[16];
#pragma unroll
  for (int e = 0; e < 16; ++e) {
    const float v = z_pre[b * FEATZ_ + e * 32 + rem];
    zf[e] = silu_f(v);
  }
  float best = 3.402823466e38f; int bi = 0x7fffffff;
  for (int j = t; j < 1024; j += 128) {
    const float* c = cbk + j * 16;
    float d = 0.f;
#pragma unroll
    for (int e = 0; e < 16; ++e) { const float df = zf[e] - c[e]; d = fmaf(df, df, d); }
    if (d < best) { best = d; bi = j; }
  }
  __shared__ float bd[128]; __shared__ int bx[128];
  bd[t] = best; bx[t] = bi;
  __syncthreads();
  for (int o = 64; o > 0; o >>= 1) {
    if (t < o) {
      if (bd[t + o] < bd[t] || (bd[t + o] == bd[t] && bx[t + o] < bx[t])) {
        bd[t] = bd[t + o]; bx[t] = bx[t + o];
      }
    }
    __syncthreads();
  }
  if (t == 0) {
    const float* c = cbk + bx[0] * 16;
    float ls = 0.f;
#pragma unroll
    for (int e = 0; e < 16; ++e) {
      const float q = c[e];
      const int kz = e * 32 + rem;
      zqT[((kz >> 1) << 6) + (b << 1) + (kz & 1)] = q;  // straight-through value
      const float df = q - zf[e];
      ls = fmaf(df, df, ls);
    }
    qrow[r] = ls;
  }
}

// y.T = Wout(131072x512) @ zqT(512x32) via WMMA; silu + bias fused;
// writes into h channels 0..127 (phylo slots). grid 8192 m-tiles, block = 1 wave.
__global__ __launch_bounds__(32) void k_mlp_out(
    const float* __restrict__ w, const float* __restrict__ zqT,
    const float* __restrict__ bias, float* __restrict__ y)
{
  const int lane = threadIdx.x;
  const int ln = lane & 15;
  const int kk = (lane >> 4) << 1;
  const int m0 = blockIdx.x * 16;
  const float* wr = w + (size_t)(m0 + ln) * FEATZ_ + kk;
  v8f acc0 = {}, acc1 = {};
  for (int k = 0; k < FEATZ_; k += 4) {
    v2f a;
    a.x = wr[k]; a.y = wr[k + 1];
    const float* zp = zqT + (k + kk) * 32 + (ln << 1);  // (k+kk) even
    v2f b0, b1;
    b0.x = zp[0];  b0.y = zp[1];
    b1.x = zp[32]; b1.y = zp[33];
    acc0 = WMMA_F32(a, b0, acc0);
    acc1 = WMMA_F32(a, b1, acc1);
  }
  const int hi = (lane >> 4) << 3;
#pragma unroll
  for (int r = 0; r < 8; ++r) {
    const int n = m0 + hi + r;
    const float bi = bias[n];
    y[(size_t)ln * 262144 + n]        = silu_f(acc0[r] + bi);
    y[(size_t)(16 + ln) * 262144 + n] = silu_f(acc1[r] + bi);
  }
}

// q_loss = 1.25 * sum / 16384, deterministic tree reduce
__global__ __launch_bounds__(256) void k_qsum(const float* __restrict__ qrow,
                                              float* __restrict__ out)
{
  const int t = threadIdx.x;
  float s = 0.f;
  for (int i = t; i < 1024; i += 256) s += qrow[i];
  __shared__ float as[256];
  as[t] = s; __syncthreads();
  for (int o = 128; o > 0; o >>= 1) { if (t < o) as[t] += as[t + o]; __syncthreads(); }
  if (t == 0) out[0] = 1.25f * as[0] * (1.f / 16384.f);
}

extern "C" void kernel_launch(void* const* d_in, const int* in_sizes, int n_in,
                              void* d_out, int out_size, void* d_ws, size_t ws_size,
                              hipStream_t stream)
{
  const float* x          = (const float*)d_in[0];
  const float* conv_in_w  = (const float*)d_in[1];
  const float* conv_in_b  = (const float*)d_in[2];
  const float* ln_w       = (const float*)d_in[3];
  const float* ln_b       = (const float*)d_in[4];
  const float* mlp_in_w   = (const float*)d_in[5];
  const float* mlp_in_b   = (const float*)d_in[6];
  const float* mlp_out_w  = (const float*)d_in[7];
  const float* mlp_out_b  = (const float*)d_in[8];
  const float* codebook   = (const float*)d_in[9];
  const float* conv_out_w = (const float*)d_in[10];
  const float* conv_out_b = (const float*)d_in[11];
  float* out = (float*)d_out;

  char* ws = (char*)d_ws;
  float* h      = (float*)ws; ws += (size_t)32 * 256 * 1024 * 4;   // 32 MB activations
  float* hnT    = (float*)ws; ws += (size_t)FEATIN_ * 32 * 4;      // 16 MB hn (interleaved)
  float* mu     = (float*)ws; ws += 256;
  float* rs     = (float*)ws; ws += 256;
  float* z_part = (float*)ws; ws += (size_t)64 * 16384 * 4;        // 4 MB split-K partials
  float* z_pre  = (float*)ws; ws += 16384 * 4;
  float* zqT    = (float*)ws; ws += 16384 * 4;
  float* qrow   = (float*)ws; ws += 1024 * 4;

  k_conv1x1_silu<<<dim3(16, 32), 256, 0, stream>>>(x, conv_in_w, conv_in_b, h);
  k_ln_stats<<<32, 256, 0, stream>>>(h, mu, rs);
  k_ln_apply<<<16384, 256, 0, stream>>>(h, ln_w, ln_b, mu, rs, hnT);
  k_mlp_in<<<dim3(32, 64), 32, 0, stream>>>(mlp_in_w, hnT, z_part);
  k_reduce_z<<<64, 256, 0, stream>>>(z_part, mlp_in_b, z_pre);
  k_vq<<<1024, 128, 0, stream>>>(z_pre, codebook, zqT, qrow);
  k_mlp_out<<<8192, 32, 0, stream>>>(mlp_out_w, zqT, mlp_out_b, h);  // overwrites phylo chans
  k_conv1x1_silu<<<dim3(16, 32), 256, 0, stream>>>(h, conv_out_w, conv_out_b, out);
  k_qsum<<<1, 256, 0, stream>>>(qrow, out + (size_t)32 * 256 * 1024);
}